// PatchShuffleModule_180388626438
// MI455X (gfx1250) — compile-verified
//
#include <hip/hip_runtime.h>
#include <stdint.h>

// ---------------------------------------------------------------------------
// PatchShuffle for x:(32,3,512,512) f32, patch 16x16 -> 32x32=1024 patches.
// Pure gather-copy: out_patch[b][i] = in_patch[b][perm_b[i]].
// Bandwidth-bound: ~201 MB traffic, ~8.6us at 23.3 TB/s roofline.
// CDNA5 path: async global->LDS->global B128 copies (ASYNCcnt), data never
// touches VGPRs. 128B (2 patch rows) per lane per fence phase.
// ---------------------------------------------------------------------------

#define AS1 __attribute__((address_space(1)))
#define AS3 __attribute__((address_space(3)))

typedef int v4i __attribute__((vector_size(16)));   // int4, matches builtin sig

#if __has_builtin(__builtin_amdgcn_global_load_async_to_lds_b128) && \
    __has_builtin(__builtin_amdgcn_global_store_async_from_lds_b128)
#define USE_ASYNC_LDS 1
#else
#define USE_ASYNC_LDS 0
#endif

__device__ __forceinline__ void wait_async0() {
#if __has_builtin(__builtin_amdgcn_s_wait_asynccnt)
    __builtin_amdgcn_s_wait_asynccnt(0);
#else
    asm volatile("s_wait_asynccnt 0" ::: "memory");
#endif
}

// 4-round Feistel bijection on 10 bits (5|5): valid permutation of [0,1024)
// for any key. Deterministic across calls (keyed only by batch index).
__device__ __forceinline__ uint32_t feistel_perm10(uint32_t p, uint32_t key) {
    uint32_t l = p >> 5, r = p & 31u;
#pragma unroll
    for (int i = 0; i < 4; ++i) {
        uint32_t f = r + key + (uint32_t)i * 0x9E3779B9u;
        f ^= f >> 16; f *= 0x7FEB352Du;
        f ^= f >> 15; f *= 0x846CA68Bu;
        f ^= f >> 16;
        uint32_t nl = r;
        r = l ^ (f & 31u);
        l = nl;
    }
    return (l << 5) | r;
}

namespace {
constexpr int B  = 32, C = 3, H = 512, W = 512, P = 16;
constexpr int NH = H / P;                   // 32
constexpr int NP = NH * NH;                 // 1024 patches
constexpr int PAIRS = B * C * NP * (P / 2); // 786,432 row-pairs of 128B
constexpr int ROW_BYTES  = P * 4;           // 64
constexpr int LINE_BYTES = W * 4;           // 2048
}

__global__ __launch_bounds__(256) void
patch_shuffle_kernel(const float* __restrict__ x, float* __restrict__ out) {
    // 128B staging slot per thread (32KB/block); each lane only touches its
    // own slot via its own async ops -> ASYNCcnt ordering suffices, no barriers.
    __shared__ float4 stage[256 * 8];
    const int tid = threadIdx.x;
#if USE_ASYNC_LDS
    AS3 v4i* slot = (AS3 v4i*)&stage[tid * 8];   // slot[0..3]=row0, [4..7]=row1
#else
    (void)stage;
#endif

    const unsigned nthreads = gridDim.x * blockDim.x;
    for (unsigned t = blockIdx.x * blockDim.x + tid; t < (unsigned)PAIRS; t += nthreads) {
        // layout: t = ((b*C + c) * NP + p_out) * 8 + rr   (NP*8 = 8192 = 2^13)
        const uint32_t rr    = t & 7u;              // row pair -> rows 2rr,2rr+1
        const uint32_t p_out = (t >> 3) & (NP - 1);
        const uint32_t bc    = t >> 13;             // b*C + c in [0,96)
        const uint32_t b     = bc / 3u;
        const uint32_t r0    = rr * 2u;

        // gather: out patch p_out <- src patch perm_b(p_out)
        const uint32_t p_src = feistel_perm10(p_out, 0x2Au * 0x01000193u + b * 0x9E3779B9u);
        const uint32_t sph = p_src >> 5, spw = p_src & 31u;
        const uint32_t dph = p_out >> 5, dpw = p_out & 31u;

        const uint32_t src_off =
            (bc * (uint32_t)H + sph * P + r0) * LINE_BYTES + spw * ROW_BYTES;
        const uint32_t dst_off =
            (bc * (uint32_t)H + dph * P + r0) * LINE_BYTES + dpw * ROW_BYTES;

#if USE_ASYNC_LDS
        AS1 v4i* gs0 = (AS1 v4i*)((const char*)x + src_off);               // row 2rr
        AS1 v4i* gs1 = (AS1 v4i*)((const char*)x + src_off + LINE_BYTES);  // row 2rr+1
        AS1 v4i* gd0 = (AS1 v4i*)((char*)out + dst_off);
        AS1 v4i* gd1 = (AS1 v4i*)((char*)out + dst_off + LINE_BYTES);
        // 2 x 64B patch rows = 8 x B128; imm offset advances global & LDS alike
        __builtin_amdgcn_global_load_async_to_lds_b128(gs0, slot,     0,  0);
        __builtin_amdgcn_global_load_async_to_lds_b128(gs0, slot,     16, 0);
        __builtin_amdgcn_global_load_async_to_lds_b128(gs0, slot,     32, 0);
        __builtin_amdgcn_global_load_async_to_lds_b128(gs0, slot,     48, 0);
        __builtin_amdgcn_global_load_async_to_lds_b128(gs1, slot + 4, 0,  0);
        __builtin_amdgcn_global_load_async_to_lds_b128(gs1, slot + 4, 16, 0);
        __builtin_amdgcn_global_load_async_to_lds_b128(gs1, slot + 4, 32, 0);
        __builtin_amdgcn_global_load_async_to_lds_b128(gs1, slot + 4, 48, 0);
        wait_async0();   // LDS slot filled (ASYNCcnt==0)
        __builtin_amdgcn_global_store_async_from_lds_b128(gd0, slot,     0,  0);
        __builtin_amdgcn_global_store_async_from_lds_b128(gd0, slot,     16, 0);
        __builtin_amdgcn_global_store_async_from_lds_b128(gd0, slot,     32, 0);
        __builtin_amdgcn_global_store_async_from_lds_b128(gd0, slot,     48, 0);
        __builtin_amdgcn_global_store_async_from_lds_b128(gd1, slot + 4, 0,  0);
        __builtin_amdgcn_global_store_async_from_lds_b128(gd1, slot + 4, 16, 0);
        __builtin_amdgcn_global_store_async_from_lds_b128(gd1, slot + 4, 32, 0);
        __builtin_amdgcn_global_store_async_from_lds_b128(gd1, slot + 4, 48, 0);
        wait_async0();   // LDS slot drained before reuse next iteration
#else
        const float4* s0 = (const float4*)((const char*)x + src_off);
        const float4* s1 = (const float4*)((const char*)x + src_off + LINE_BYTES);
        float4* d0 = (float4*)((char*)out + dst_off);
        float4* d1 = (float4*)((char*)out + dst_off + LINE_BYTES);
        float4 a0 = s0[0], a1 = s0[1], a2 = s0[2], a3 = s0[3];
        float4 b0 = s1[0], b1 = s1[1], b2 = s1[2], b3 = s1[3];
        d0[0] = a0; d0[1] = a1; d0[2] = a2; d0[3] = a3;
        d1[0] = b0; d1[1] = b1; d1[2] = b2; d1[3] = b3;
#endif
    }
}

extern "C" void kernel_launch(void* const* d_in, const int* in_sizes, int n_in,
                              void* d_out, int out_size, void* d_ws, size_t ws_size,
                              hipStream_t stream) {
    (void)in_sizes; (void)n_in; (void)out_size; (void)d_ws; (void)ws_size;
    const float* x   = (const float*)d_in[0];   // (32,3,512,512) f32
    float*       out = (float*)d_out;
    // 1536 blocks x 256 threads = 393,216 threads; PAIRS/threads = 2 uniform
    // iterations each (no tail divergence).
    patch_shuffle_kernel<<<dim3(1536), dim3(256), 0, stream>>>(x, out);
}